// NeuralSemiLagrangian_26036091748470
// MI455X (gfx1250) — compile-verified
//
#include <hip/hip_runtime.h>
#include <hip/hip_bf16.h>

typedef __attribute__((ext_vector_type(2))) float v2f;
typedef __attribute__((ext_vector_type(8))) float v8f;

union V8 { v8f v; float f[8]; };

constexpr int Cc = 64;
constexpr int Hh = 721;
constexpr int Ww = 1440;
constexpr int NV = 8;
constexpr int HW = Hh * Ww;          // 1,038,240 (divisible by 16)
constexpr int Hp = Hh + 2;           // 723
constexpr int Wp = Ww + 2;           // 1442
constexpr int NTILES = HW / 16;      // 64,890 wave-tiles

// ---------------------------------------------------------------------------
// Kernel A: per-pixel LayerNorm statistics + down-projection into padded buf
// ---------------------------------------------------------------------------
__global__ void __launch_bounds__(256) k_norm_proj(
    const float* __restrict__ hid,
    const float* __restrict__ dwn_w, const float* __restrict__ dwn_b,
    float* __restrict__ mu, float* __restrict__ rv, float* __restrict__ pp)
{
    int pix = blockIdx.x * blockDim.x + threadIdx.x;
    if (pix >= HW) return;
    float s = 0.f, s2 = 0.f;
    float p[NV];
#pragma unroll
    for (int o = 0; o < NV; o++) p[o] = 0.f;
#pragma unroll 4
    for (int c = 0; c < Cc; c++) {
        if ((c & 15) == 0 && c + 16 < Cc)
            __builtin_prefetch(&hid[(size_t)(c + 16) * HW + pix], 0, 0);
        float x = hid[(size_t)c * HW + pix];
        s += x; s2 = fmaf(x, x, s2);
#pragma unroll
        for (int o = 0; o < NV; o++) p[o] = fmaf(dwn_w[o * Cc + c], x, p[o]);
    }
    float m   = s * (1.f / Cc);
    float var = fmaf(-m, m, s2 * (1.f / Cc));
    mu[pix] = m;
    rv[pix] = rsqrtf(var + 1e-5f);
    int h = pix / Ww, w = pix - h * Ww;
    size_t base = (size_t)(h + 1) * Wp + (w + 1);
#pragma unroll
    for (int o = 0; o < NV; o++)
        pp[(size_t)o * Hp * Wp + base] = p[o] + dwn_b[o];
}

// Halo fill: rows first (polar rows = row 0 / H-1 rolled by W/2) ...
__global__ void k_halo_rows(float* __restrict__ pp)
{
    int t = blockIdx.x * blockDim.x + threadIdx.x;
    if (t >= NV * Ww) return;
    int nv = t / Ww, w = t - nv * Ww;
    int ws = (w + Ww / 2) % Ww;
    float* plane = pp + (size_t)nv * Hp * Wp;
    plane[(size_t)0 * Wp + (w + 1)]        = plane[(size_t)1 * Wp + (ws + 1)];
    plane[(size_t)(Hp - 1) * Wp + (w + 1)] = plane[(size_t)(Hp - 2) * Wp + (ws + 1)];
}
// ... then cyclic columns of the row-padded array.
__global__ void k_halo_cols(float* __restrict__ pp)
{
    int t = blockIdx.x * blockDim.x + threadIdx.x;
    if (t >= NV * Hp) return;
    int nv = t / Hp, r = t - nv * Hp;
    float* row = pp + (size_t)nv * Hp * Wp + (size_t)r * Wp;
    row[0]      = row[Wp - 2];
    row[Wp - 1] = row[1];
}

// ---------------------------------------------------------------------------
// Kernel B: LN+depthwise on the fly, 16x64 pointwise via WMMA f32 16x16x4,
// then departure-point coordinates gx/gy.
// Wave tile = 16 pixels; lane = (half, pixel): half selects K-pair / u-vs-v.
// ---------------------------------------------------------------------------
__global__ void __launch_bounds__(256) k_velocity(
    const float* __restrict__ hid, const float* __restrict__ mu,
    const float* __restrict__ rv,
    const float* __restrict__ latg, const float* __restrict__ long_,
    const float* __restrict__ dtp,
    const float* __restrict__ gamma, const float* __restrict__ beta,
    const float* __restrict__ dww, const float* __restrict__ dwb,
    const float* __restrict__ pww, const float* __restrict__ pwb,
    float* __restrict__ gxb, float* __restrict__ gyb)
{
    int tile = blockIdx.x * (blockDim.x >> 5) + (threadIdx.x >> 5);
    if (tile >= NTILES) return;                 // uniform per wave: EXEC stays all-1s
    int lane = threadIdx.x & 31;
    int half = lane >> 4;
    int col  = lane & 15;
    int pix  = tile * 16 + col;                 // W%16==0 -> tile stays in one row
    int h = pix / Ww, w = pix - h * Ww;

    // 9 geo-cyclic neighbor offsets + LN stats (shared across all channels)
    int off[9]; float m9[9], r9[9];
#pragma unroll
    for (int j = 0; j < 9; j++) {
        int dy = j / 3 - 1, dx = (j - (j / 3) * 3) - 1;
        int ww = w + dx;
        if (ww < 0)    ww += Ww;
        if (ww >= Ww)  ww -= Ww;
        int hh = h + dy;
        if (hh < 0)   { hh = 0;      ww = (ww + Ww / 2) % Ww; }
        if (hh >= Hh) { hh = Hh - 1; ww = (ww + Ww / 2) % Ww; }
        off[j] = hh * Ww + ww;
        m9[j] = mu[off[j]];
        r9[j] = rv[off[j]];
    }

    v8f acc = {};
    for (int k = 0; k < 16; k++) {              // K = 64 in steps of 4
        int cb = 4 * k + 2 * half;
        v2f a, b;
#pragma unroll
        for (int e = 0; e < 2; e++) {
            int c = cb + e;
            float g = gamma[c], be = beta[c], d = dwb[c];
#pragma unroll
            for (int j = 0; j < 9; j++) {
                float xn = fmaf((hid[(size_t)c * HW + off[j]] - m9[j]) * r9[j], g, be);
                d = fmaf(dww[c * 9 + j], xn, d);
            }
            b[e] = d;                           // B: K=cb+e, N=col
            a[e] = pww[col * Cc + c];           // A: M=col,  K=cb+e
        }
        acc = __builtin_amdgcn_wmma_f32_16x16x4_f32(false, a, false, b,
                                                    (short)0, acc, false, false);
    }

    float lat = latg[pix], lon = long_[pix];
    float dt = dtp[0];
    float mnlat = latg[0], mxlat = latg[(size_t)(Hh - 1) * Ww];
    float mnlon = long_[0], mxlon = long_[Ww - 1];
    float rcos = 1.f / __cosf(lat);
    float sclx = 2.f / (mxlon - mnlon), scly = 2.f / (mxlat - mnlat);

    V8 av; av.v = acc;                          // VGPR j: rows j (u, half0) / j+8 (v, half1)
#pragma unroll
    for (int j = 0; j < NV; j++) {
        float t = av.f[j] + pwb[j + 8 * half];
        // u-path (valid on half0): gx with cyclic wrap
        float lon_dep = lon - (t * rcos) * dt;
        float gx = fmaf(lon_dep - mnlon, sclx, -1.f);
        float g1 = gx + 1.f;
        gx = (g1 - 2.f * floorf(g1 * 0.5f)) - 1.f;     // remainder(gx+1,2)-1
        // v-path (valid on half1): gy before pole reflection
        float lat_dep = lat - t * dt;
        float gy = fmaf(lat_dep - mnlat, scly, -1.f);
        float gy_partner = __shfl_xor(gy, 16, 32);     // hand v-lane's gy to u-lane
        if (half == 0) {
            if (fabsf(gy_partner) > 1.f) {             // sequential where-semantics
                if (gx <= 0.f) gx += 1.f;
                if (gx >  0.f) gx -= 1.f;
            }
            gxb[(size_t)j * HW + pix] = gx * ((float)Ww / (float)Wp);
        } else {
            float gyc = gy;                            // sequential reflection
            if (gyc < -1.f) gyc = -(2.f + gyc);
            if (gyc >  1.f) gyc =  2.f - gyc;
            gyb[(size_t)j * HW + pix] = gyc * ((float)Hh / (float)Hp);
        }
    }
}

// ---------------------------------------------------------------------------
// Kernel C: bicubic gather from padded proj + 64x8 up-projection via WMMA.
// ---------------------------------------------------------------------------
__device__ __forceinline__ void cubw(float t, float w[4])
{
    const float a = -0.75f;
    float t1 = t + 1.f, u = 1.f - t, v = 2.f - t;
    w[0] = ((a * t1 - 5.f * a) * t1 + 8.f * a) * t1 - 4.f * a;
    w[1] = ((a + 2.f) * t - (a + 3.f)) * t * t + 1.f;
    w[2] = ((a + 2.f) * u - (a + 3.f)) * u * u + 1.f;
    w[3] = ((a * v - 5.f * a) * v + 8.f * a) * v - 4.f * a;
}

__device__ __forceinline__ float bicubic(const float* __restrict__ img,
                                         float gx, float gy)
{
    float ix = (gx + 1.f) * 0.5f * (float)(Wp - 1);
    float iy = (gy + 1.f) * 0.5f * (float)(Hp - 1);
    float fx0 = floorf(ix), fy0 = floorf(iy);
    float wx[4], wy[4];
    cubw(ix - fx0, wx);
    cubw(iy - fy0, wy);
    int ix0 = (int)fx0, iy0 = (int)fy0;
    float out = 0.f;
#pragma unroll
    for (int m = 0; m < 4; m++) {
        int yi = iy0 + m - 1;
        yi = yi < 0 ? 0 : (yi > Hp - 1 ? Hp - 1 : yi);
        const float* rowp = img + (size_t)yi * Wp;
        float row = 0.f;
#pragma unroll
        for (int n = 0; n < 4; n++) {
            int xi = ix0 + n - 1;
            xi = xi < 0 ? 0 : (xi > Wp - 1 ? Wp - 1 : xi);
            row = fmaf(wx[n], rowp[xi], row);
        }
        out = fmaf(wy[m], row, out);
    }
    return out;
}

__global__ void __launch_bounds__(256) k_advect_up(
    const float* __restrict__ pp,
    const float* __restrict__ gxb, const float* __restrict__ gyb,
    const float* __restrict__ upw, const float* __restrict__ upb,
    float* __restrict__ out)
{
    int tile = blockIdx.x * (blockDim.x >> 5) + (threadIdx.x >> 5);
    if (tile >= NTILES) return;
    int lane = threadIdx.x & 31;
    int half = lane >> 4;
    int col  = lane & 15;
    int pix  = tile * 16 + col;

    // B fragments: interp rows nv = 4s + 2*half + e at column `col`
    v2f bfr[2];
#pragma unroll
    for (int s = 0; s < 2; s++) {
#pragma unroll
        for (int e = 0; e < 2; e++) {
            int nv = 4 * s + 2 * half + e;
            float gx = gxb[(size_t)nv * HW + pix];
            float gy = gyb[(size_t)nv * HW + pix];
            bfr[s][e] = bicubic(pp + (size_t)nv * Hp * Wp, gx, gy);
        }
    }

    v8f acc[4] = { {}, {}, {}, {} };            // 4 M-tiles of 16 channels
#pragma unroll
    for (int s = 0; s < 2; s++) {               // K = 8 in steps of 4
#pragma unroll
        for (int m = 0; m < 4; m++) {
            v2f a;
            int row = 16 * m + col;
            a[0] = upw[row * NV + 4 * s + 2 * half];
            a[1] = upw[row * NV + 4 * s + 2 * half + 1];
            acc[m] = __builtin_amdgcn_wmma_f32_16x16x4_f32(false, a, false, bfr[s],
                                                           (short)0, acc[m], false, false);
        }
    }
#pragma unroll
    for (int m = 0; m < 4; m++) {
        V8 av; av.v = acc[m];
#pragma unroll
        for (int j = 0; j < 8; j++) {
            int c = 16 * m + j + 8 * half;
            out[(size_t)c * HW + pix] = av.f[j] + upb[c];
        }
    }
}

// ---------------------------------------------------------------------------
extern "C" void kernel_launch(void* const* d_in, const int* in_sizes, int n_in,
                              void* d_out, int out_size, void* d_ws, size_t ws_size,
                              hipStream_t stream)
{
    (void)in_sizes; (void)n_in; (void)out_size; (void)ws_size;
    const float* hid   = (const float*)d_in[0];
    const float* latg  = (const float*)d_in[1];
    const float* lon   = (const float*)d_in[2];
    const float* dt    = (const float*)d_in[3];
    const float* dwn_w = (const float*)d_in[4];
    const float* dwn_b = (const float*)d_in[5];
    const float* up_w  = (const float*)d_in[6];
    const float* up_b  = (const float*)d_in[7];
    const float* gam   = (const float*)d_in[8];
    const float* bet   = (const float*)d_in[9];
    const float* dww   = (const float*)d_in[10];
    const float* dwb   = (const float*)d_in[11];
    const float* pww   = (const float*)d_in[12];
    const float* pwb   = (const float*)d_in[13];
    float* out = (float*)d_out;

    float* ws = (float*)d_ws;
    float* mu = ws;                                   // HW
    float* rv = ws + (size_t)HW;                      // HW
    float* gx = ws + (size_t)2 * HW;                  // 8*HW
    float* gy = ws + (size_t)10 * HW;                 // 8*HW
    float* pp = ws + (size_t)18 * HW;                 // 8*Hp*Wp

    k_norm_proj<<<(HW + 255) / 256, 256, 0, stream>>>(hid, dwn_w, dwn_b, mu, rv, pp);
    k_halo_rows<<<(NV * Ww + 255) / 256, 256, 0, stream>>>(pp);
    k_halo_cols<<<(NV * Hp + 255) / 256, 256, 0, stream>>>(pp);
    k_velocity<<<(NTILES * 32 + 255) / 256, 256, 0, stream>>>(
        hid, mu, rv, latg, lon, dt, gam, bet, dww, dwb, pww, pwb, gx, gy);
    k_advect_up<<<(NTILES * 32 + 255) / 256, 256, 0, stream>>>(
        pp, gx, gy, up_w, up_b, out);
}